// GCNConv_15779709845545
// MI455X (gfx1250) — compile-verified
//
#include <hip/hip_runtime.h>

#define N_NODES 50000
#define N_EDGES 800000
#define D 64  // D_IN == D_OUT == 64

typedef __attribute__((ext_vector_type(2))) float v2f;
typedef __attribute__((ext_vector_type(8))) float v8f;

// ---------------------------------------------------------------------------
// Kernel 1: zero the workspace (sum[N*64] followed by deg[N])
// ---------------------------------------------------------------------------
__global__ __launch_bounds__(256) void zero_ws_kernel(float* __restrict__ p, int n) {
    int i = blockIdx.x * 256 + threadIdx.x;
    if (i < n) p[i] = 0.0f;
}

// ---------------------------------------------------------------------------
// Kernel 2: edge scatter-add. threadIdx.x = feature lane (0..63) so each
// wave32 issues 32 consecutive-dword atomics into one destination row
// (2 cachelines). sum/deg live in L2 (13 MB << 192 MB L2), so these are
// L2-resident global_atomic_add_f32 ops.
// ---------------------------------------------------------------------------
__global__ __launch_bounds__(256) void scatter_accum_kernel(
    const float* __restrict__ x, const int* __restrict__ src,
    const int* __restrict__ dst, float* __restrict__ sum,
    float* __restrict__ deg) {
    const int tx = threadIdx.x;                 // feature 0..63
    const int e  = blockIdx.x * 4 + threadIdx.y;
    if (e >= N_EDGES) return;
    const int s = src[e];
    const int d = dst[e];
    const float v = x[(size_t)s * D + tx];
    __hip_atomic_fetch_add(&sum[(size_t)d * D + tx], v,
                           __ATOMIC_RELAXED, __HIP_MEMORY_SCOPE_AGENT);
    if (tx == 0) {
        __hip_atomic_fetch_add(&deg[d], 1.0f,
                               __ATOMIC_RELAXED, __HIP_MEMORY_SCOPE_AGENT);
    }
}

// ---------------------------------------------------------------------------
// Kernel 3: out = (sum / max(deg,1)) @ W^T + b  via V_WMMA_F32_16X16X4_F32.
// Block = 128 threads = 4 waves; block owns 16 output rows, each wave owns a
// 16-wide slice of the 64 output cols. K=64 -> 16 chained f32 WMMAs.
//
// f32 16x4 A layout (ISA 7.12.2): lane<16 holds M=lane, K={k0,k0+1};
// lane>=16 holds M=lane-16, K={k0+2,k0+3}. B (4x16, K-major rows striped
// across lanes) mirrors this with N in place of M. Both fragments are two
// consecutive floats -> single b64 loads. Degree normalization is folded
// into the A fragment; bias into the store. C/D layout: VGPR i holds
// M = i + 8*(lane>=16), N = lane&15.
// ---------------------------------------------------------------------------
__global__ __launch_bounds__(128) void gemm_wmma_kernel(
    const float* __restrict__ sum, const float* __restrict__ deg,
    const float* __restrict__ W, const float* __restrict__ bias,
    float* __restrict__ out) {
    const int lane = threadIdx.x & 31;
    const int wave = threadIdx.x >> 5;
    const int half = lane >> 4;   // 0: K pair {0,1}; 1: K pair {2,3}
    const int r    = lane & 15;   // row (A) / col (B) within tile
    const int m0   = blockIdx.x * 16;
    const int n0   = wave * 16;

    // per-row 1/max(deg,1): lane's A-rows are all m0+r
    const float rd = 1.0f / fmaxf(deg[m0 + r], 1.0f);

    const float* __restrict__ arow = sum + (size_t)(m0 + r) * D;
    const float* __restrict__ brow = W   + (size_t)(n0 + r) * D;  // B[k][n] = W[n][k]

    v8f c = {};
#pragma unroll
    for (int k0 = 0; k0 < D; k0 += 4) {
        const int kk = k0 + 2 * half;
        v2f a = *(const v2f*)(arow + kk);
        v2f b = *(const v2f*)(brow + kk);
        a.x *= rd;
        a.y *= rd;
        c = __builtin_amdgcn_wmma_f32_16x16x4_f32(
                /*neg_a=*/false, a, /*neg_b=*/false, b,
                /*c_mod=*/(short)0, c, /*reuse_a=*/false, /*reuse_b=*/false);
    }

    const float bv = bias[n0 + r];
#pragma unroll
    for (int i = 0; i < 8; ++i) {
        out[(size_t)(m0 + i + 8 * half) * D + (n0 + r)] = c[i] + bv;
    }
}

// ---------------------------------------------------------------------------
extern "C" void kernel_launch(void* const* d_in, const int* in_sizes, int n_in,
                              void* d_out, int out_size, void* d_ws, size_t ws_size,
                              hipStream_t stream) {
    const float* x    = (const float*)d_in[0];
    const int*   src  = (const int*)d_in[1];
    const int*   dst  = (const int*)d_in[2];
    const float* W    = (const float*)d_in[3];
    const float* bias = (const float*)d_in[4];
    float*       out  = (float*)d_out;

    float* sum = (float*)d_ws;                      // [N_NODES * D]
    float* deg = sum + (size_t)N_NODES * D;         // [N_NODES]

    // 1) zero workspace
    const int zn = N_NODES * D + N_NODES;
    zero_ws_kernel<<<(zn + 255) / 256, 256, 0, stream>>>((float*)d_ws, zn);

    // 2) scatter-add messages + degrees
    dim3 sblk(64, 4);
    scatter_accum_kernel<<<(N_EDGES + 3) / 4, sblk, 0, stream>>>(x, src, dst, sum, deg);

    // 3) mean-normalize + dense linear via f32 WMMA
    gemm_wmma_kernel<<<N_NODES / 16, 128, 0, stream>>>(sum, deg, W, bias, out);
}